// PointNetFeaturePropagation_2508260901535
// MI455X (gfx1250) — compile-verified
//
#include <hip/hip_runtime.h>

typedef __attribute__((ext_vector_type(2))) float v2f;
typedef __attribute__((ext_vector_type(8))) float v8f;

#define BB   2
#define NN   16384
#define SS   4096
#define DD   256
#define NT   (BB * NN)      /* 32768 GEMM columns (both batches fused) */
#define C0   512            /* layer-0 K */
#define O0   256
#define O1   256

// ---------------------------------------------------------------------------
// Kernel 1: brute-force 3-NN + inverse-distance weights.
// xyz2 (3 x 4096 per batch) is staged in LDS (48KB of the 320KB/WGP).
// ---------------------------------------------------------------------------
__global__ __launch_bounds__(256) void k_three_nn(
    const float* __restrict__ xyz1, const float* __restrict__ xyz2,
    float* __restrict__ wgt, int* __restrict__ idx)
{
  __shared__ float lx[SS], ly[SS], lz[SS];
  const int b = blockIdx.y;
  const float* x2 = xyz2 + (long)b * 3 * SS;
  for (int s = threadIdx.x; s < SS; s += 256) {
    lx[s] = x2[s];
    ly[s] = x2[SS + s];
    lz[s] = x2[2 * SS + s];
  }
  __syncthreads();

  const int n = blockIdx.x * 256 + threadIdx.x;
  const float* x1 = xyz1 + (long)b * 3 * NN;
  const float px = x1[n], py = x1[NN + n], pz = x1[2 * NN + n];

  float b0 = 3.4e38f, b1 = 3.4e38f, b2 = 3.4e38f;
  int   i0 = 0, i1 = 0, i2 = 0;
  for (int s = 0; s < SS; ++s) {
    const float dx = px - lx[s], dy = py - ly[s], dz = pz - lz[s];
    const float d = dx * dx + dy * dy + dz * dz;
    if (d < b2) {
      if (d < b1) {
        if (d < b0) { b2 = b1; i2 = i1; b1 = b0; i1 = i0; b0 = d; i0 = s; }
        else        { b2 = b1; i2 = i1; b1 = d;  i1 = s; }
      } else        { b2 = d;  i2 = s; }
    }
  }
  const float w0 = 1.f / (b0 + 1e-8f);
  const float w1 = 1.f / (b1 + 1e-8f);
  const float w2 = 1.f / (b2 + 1e-8f);
  const float inv = 1.f / (w0 + w1 + w2);
  const long p = (long)(b * NN + n) * 3;
  wgt[p] = w0 * inv; wgt[p + 1] = w1 * inv; wgt[p + 2] = w2 * inv;
  idx[p] = i0;       idx[p + 1] = i1;       idx[p + 2] = i2;
}

// ---------------------------------------------------------------------------
// Kernel 2: transpose points2 (b,c,s) -> f2t (b,s,c) so feature gathers are
// contiguous 256-float rows (all L2-resident: 8.4MB << 192MB L2).
// ---------------------------------------------------------------------------
__global__ __launch_bounds__(256) void k_transpose_f2(
    const float* __restrict__ p2, float* __restrict__ f2t)
{
  __shared__ float tile[16][17];
  const int b  = blockIdx.z;
  const int c0 = blockIdx.x * 16;
  const int s0 = blockIdx.y * 16;
  const int tx = threadIdx.x & 15;
  const int ty = threadIdx.x >> 4;
  tile[ty][tx] = p2[(long)b * DD * SS + (long)(c0 + ty) * SS + (s0 + tx)];
  __syncthreads();
  f2t[(long)b * SS * DD + (long)(s0 + ty) * DD + (c0 + tx)] = tile[tx][ty];
}

// ---------------------------------------------------------------------------
// Kernel 3: build X0 (512 rows x NT cols, n-contiguous = WMMA B layout).
// Rows [0,256)   : copy of points1.
// Rows [256,512) : 3-NN weighted interpolation of f2t.
// ---------------------------------------------------------------------------
__global__ __launch_bounds__(256) void k_build_x0(
    const float* __restrict__ p1, const float* __restrict__ f2t,
    const float* __restrict__ wgt, const int* __restrict__ idx,
    float* __restrict__ X0)
{
  const int b = blockIdx.y;
  const int n = blockIdx.x * 256 + threadIdx.x;
  const long col = (long)b * NN + n;
  const long p = col * 3;
  const float w0 = wgt[p], w1 = wgt[p + 1], w2 = wgt[p + 2];
  const float* f0 = f2t + ((long)b * SS + idx[p])     * DD;
  const float* f1 = f2t + ((long)b * SS + idx[p + 1]) * DD;
  const float* f2 = f2t + ((long)b * SS + idx[p + 2]) * DD;
  const float* src = p1 + (long)b * DD * NN + n;
  for (int c = 0; c < DD; ++c) {
    X0[(long)c * NT + col]        = src[(long)c * NN];
    X0[(long)(DD + c) * NT + col] = w0 * f0[c] + w1 * f1[c] + w2 * f2[c];
  }
}

// ---------------------------------------------------------------------------
// Kernel 4: Z = W * X + bias via V_WMMA_F32_16X16X4_F32.
// Each wave owns a 16(o) x 64(n) C tile: 4 x v8f accumulators, so one A
// (weight) fragment feeds 4 WMMAs -> 4x less A traffic, WMMA:VMEM issue
// ratio 4:9 instead of 1:3.
// A fragment (16x4 f32, 2 VGPRs): lanes 0-15 -> K=kb+0/1, lanes 16-31 -> K=kb+2/3.
// B fragment (4x16 f32, 2 VGPRs): mirrored striping over N=lane%16.
// C/D (8 VGPRs): M = r + 8*(lane>=16), N = lane%16 (ISA 7.12.2).
// ---------------------------------------------------------------------------
__global__ __launch_bounds__(128) void k_gemm_bias(
    const float* __restrict__ Wm, const float* __restrict__ bias,
    const float* __restrict__ X, float* __restrict__ Z, int K)
{
  const int lane = threadIdx.x & 31;
  const int wave = threadIdx.x >> 5;
  const int o0   = blockIdx.y * 16;
  const long n0  = (long)blockIdx.x * 256 + (long)wave * 64;
  const int m    = lane & 15;            // A row / C column-lane id
  const int kk   = (lane >> 4) << 1;     // K sub-offset: 0 or 2

  const float* Aptr = Wm + (long)(o0 + m) * K + kk;
  const float* Bptr = X + (long)kk * NT + n0 + m;

  v8f acc0 = {}, acc1 = {}, acc2 = {}, acc3 = {};
  #pragma unroll 2
  for (int kb = 0; kb < K; kb += 4) {
    const v2f a = *(const v2f*)Aptr;     // A[m][kb+kk], A[m][kb+kk+1]
    v2f bf0, bf1, bf2, bf3;
    bf0.x = Bptr[0];       bf0.y = Bptr[NT];        // n subtile 0
    bf1.x = Bptr[16];      bf1.y = Bptr[NT + 16];   // n subtile 1
    bf2.x = Bptr[32];      bf2.y = Bptr[NT + 32];   // n subtile 2
    bf3.x = Bptr[48];      bf3.y = Bptr[NT + 48];   // n subtile 3
    acc0 = __builtin_amdgcn_wmma_f32_16x16x4_f32(false, a, false, bf0,
                                                 (short)0, acc0, false, false);
    acc1 = __builtin_amdgcn_wmma_f32_16x16x4_f32(false, a, false, bf1,
                                                 (short)0, acc1, false, false);
    acc2 = __builtin_amdgcn_wmma_f32_16x16x4_f32(false, a, false, bf2,
                                                 (short)0, acc2, false, false);
    acc3 = __builtin_amdgcn_wmma_f32_16x16x4_f32(false, a, false, bf3,
                                                 (short)0, acc3, false, false);
    Aptr += 4;
    Bptr += (long)4 * NT;
  }

  const int mhalf = (lane >> 4) << 3;    // +8 rows for upper half-wave
  #pragma unroll
  for (int r = 0; r < 8; ++r) {
    const int row = o0 + r + mhalf;
    const float bv = bias[row];
    float* zr = Z + (long)row * NT + n0 + m;
    zr[0]  = acc0[r] + bv;
    zr[16] = acc1[r] + bv;
    zr[32] = acc2[r] + bv;
    zr[48] = acc3[r] + bv;
  }
}

// ---------------------------------------------------------------------------
// Kernel 5: per-channel sum / sum-of-squares (one block per channel).
// ---------------------------------------------------------------------------
__global__ __launch_bounds__(256) void k_stats(
    const float* __restrict__ Z, float* __restrict__ st)
{
  const int ch = blockIdx.x;
  const float* row = Z + (long)ch * NT;
  float s = 0.f, sq = 0.f;
  for (int i = threadIdx.x; i < NT; i += 256) {
    const float v = row[i];
    s += v; sq += v * v;
  }
  __shared__ float ss[256], ssq[256];
  ss[threadIdx.x] = s; ssq[threadIdx.x] = sq;
  __syncthreads();
  for (int off = 128; off > 0; off >>= 1) {
    if (threadIdx.x < off) {
      ss[threadIdx.x]  += ss[threadIdx.x + off];
      ssq[threadIdx.x] += ssq[threadIdx.x + off];
    }
    __syncthreads();
  }
  if (threadIdx.x == 0) { st[ch] = ss[0]; st[256 + ch] = ssq[0]; }
}

// ---------------------------------------------------------------------------
// Kernel 6: BN + ReLU. final_layout=1 writes d_out in (b, o, n) order.
// ---------------------------------------------------------------------------
__global__ __launch_bounds__(256) void k_bn_relu(
    const float* __restrict__ Z, const float* __restrict__ st,
    const float* __restrict__ g, const float* __restrict__ be,
    float* __restrict__ Y, int final_layout)
{
  const int ch  = blockIdx.y;
  const long col = (long)blockIdx.x * 256 + threadIdx.x;
  const float mean = st[ch] * (1.f / NT);
  const float var  = st[256 + ch] * (1.f / NT) - mean * mean;
  const float rs   = rsqrtf(var + 1e-5f);
  float v = (Z[(long)ch * NT + col] - mean) * rs * g[ch] + be[ch];
  v = fmaxf(v, 0.f);
  if (final_layout) {
    const int b = (int)(col >> 14);      // col / 16384
    const int n = (int)(col & 16383);
    Y[((long)b * 256 + ch) * NN + n] = v;
  } else {
    Y[(long)ch * NT + col] = v;
  }
}

// ---------------------------------------------------------------------------
extern "C" void kernel_launch(void* const* d_in, const int* in_sizes, int n_in,
                              void* d_out, int out_size, void* d_ws, size_t ws_size,
                              hipStream_t stream)
{
  (void)in_sizes; (void)n_in; (void)out_size; (void)ws_size;

  const float* xyz1 = (const float*)d_in[0];
  const float* xyz2 = (const float*)d_in[1];
  const float* pts1 = (const float*)d_in[2];
  const float* pts2 = (const float*)d_in[3];
  const float* W0   = (const float*)d_in[4];
  const float* b0   = (const float*)d_in[5];
  const float* g0   = (const float*)d_in[6];
  const float* be0  = (const float*)d_in[7];
  const float* W1   = (const float*)d_in[8];
  const float* b1   = (const float*)d_in[9];
  const float* g1   = (const float*)d_in[10];
  const float* be1  = (const float*)d_in[11];

  // Workspace layout (floats)
  float* wgt = (float*)d_ws;                       // 3*NT
  int*   idx = (int*)(wgt + 3 * NT);               // 3*NT
  float* f2t = (float*)(idx + 3 * NT);             // BB*SS*DD
  float* X0  = f2t + (long)BB * SS * DD;           // C0*NT
  float* Z   = X0 + (long)C0 * NT;                 // O0*NT (reused by layer 1)
  float* Y0  = Z + (long)O0 * NT;                  // O0*NT
  float* st  = Y0 + (long)O0 * NT;                 // 512

  // 1) 3-NN + weights
  k_three_nn<<<dim3(NN / 256, BB), 256, 0, stream>>>(xyz1, xyz2, wgt, idx);

  // 2) points2 transpose for contiguous gathers
  k_transpose_f2<<<dim3(DD / 16, SS / 16, BB), 256, 0, stream>>>(pts2, f2t);

  // 3) build X0 = concat(points1, interp)  [512 x 32768]
  k_build_x0<<<dim3(NN / 256, BB), 256, 0, stream>>>(pts1, f2t, wgt, idx, X0);

  // 4) layer 0: GEMM + bias -> stats -> BN+ReLU
  k_gemm_bias<<<dim3(NT / 256, O0 / 16), 128, 0, stream>>>(W0, b0, X0, Z, C0);
  k_stats<<<O0, 256, 0, stream>>>(Z, st);
  k_bn_relu<<<dim3(NT / 256, O0), 256, 0, stream>>>(Z, st, g0, be0, Y0, 0);

  // 5) layer 1: GEMM + bias -> stats -> BN+ReLU -> d_out (b,o,n layout)
  k_gemm_bias<<<dim3(NT / 256, O1 / 16), 128, 0, stream>>>(W1, b1, Y0, Z, O0);
  k_stats<<<O1, 256, 0, stream>>>(Z, st);
  k_bn_relu<<<dim3(NT / 256, O1), 256, 0, stream>>>(Z, st, g1, be1,
                                                    (float*)d_out, 1);
}